// GraphSAGE_26379689132538
// MI455X (gfx1250) — compile-verified
//
#include <hip/hip_runtime.h>

typedef __attribute__((ext_vector_type(2))) float v2f;
typedef __attribute__((ext_vector_type(8))) float v8f;

#define N_NODES   100000
#define N_EDGES   1600000
#define FEAT      128     // in-feats == hidden == 128 (all scatter inputs are 128-wide)

// ---------------------------------------------------------------- utilities
__global__ void fill_zero_kernel(float* __restrict__ p, int n) {
    int i = blockIdx.x * blockDim.x + threadIdx.x;
    if (i < n) p[i] = 0.0f;
}

__global__ void degree_kernel(const int* __restrict__ dst, float* __restrict__ deg) {
    int e = blockIdx.x * blockDim.x + threadIdx.x;
    if (e < N_EDGES) atomicAdd(&deg[dst[e]], 1.0f);
}

__global__ void invdeg_kernel(float* __restrict__ deg) {
    int i = blockIdx.x * blockDim.x + threadIdx.x;
    if (i < N_NODES) deg[i] = 1.0f / fmaxf(deg[i], 1.0f);
}

// ------------------------------------------------- edge scatter-add (segment_sum)
// One 32-lane group per edge; lane handles 4 consecutive floats of the 128-wide
// row -> 512B coalesced load from x[src], 4 f32 atomics into agg[dst].
__global__ void scatter_add_kernel(const float* __restrict__ x,
                                   const int* __restrict__ src,
                                   const int* __restrict__ dst,
                                   float* __restrict__ agg) {
    int t    = blockIdx.x * blockDim.x + threadIdx.x;   // < N_EDGES*32 (51.2M)
    int e    = t >> 5;
    int part = t & 31;
    if (e >= N_EDGES) return;
    int s = src[e];
    int d = dst[e];
    const float4 v = *(const float4*)(x + (size_t)s * FEAT + part * 4);
    float* ap = agg + (size_t)d * FEAT + part * 4;
    atomicAdd(ap + 0, v.x);
    atomicAdd(ap + 1, v.y);
    atomicAdd(ap + 2, v.z);
    atomicAdd(ap + 3, v.w);
}

// ------------------------------------------------- fused SAGE GEMM via f32 WMMA
// out[m, :] = act( x[m,:]@Ws + (inv_deg[m]*agg[m,:])@Wn + b )
// One wave per 16-row stripe; NT 16-col tiles (NT=8 -> 128 cols, NT=3 masked -> 40).
// A fragment (16x4 f32): lane l: row = l&15; holds K = (l>>4)*2 + {0,1}.
// B fragment (4x16 f32): lane l: col = l&15; holds K = (l>>4)*2 + {0,1}.
// C/D (16x16 f32, 8 VGPRs): vgpr r, lanes 0-15 -> M=r, lanes 16-31 -> M=8+r; N=l&15.
template <int NT, bool RELU, int NOUT>
__global__ void sage_gemm_kernel(const float* __restrict__ x,
                                 const float* __restrict__ agg,
                                 const float* __restrict__ inv_deg,
                                 const float* __restrict__ Ws,
                                 const float* __restrict__ Wn,
                                 const float* __restrict__ bias,
                                 float* __restrict__ out) {
    const int lane = threadIdx.x;            // 0..31, one wave per block
    const int m0   = blockIdx.x * 16;
    const int arow = m0 + (lane & 15);
    const int kk   = (lane >> 4) * 2;        // K sub-offset for this lane half
    const float scale = inv_deg[arow];

    // Branch-free column masking for the NOUT=40 tail tile (EXEC stays all-1s).
    int   ncol[NT];
    float nmsk[NT];
#pragma unroll
    for (int t = 0; t < NT; ++t) {
        int n   = t * 16 + (lane & 15);
        ncol[t] = (n < NOUT) ? n : (NOUT - 1);
        nmsk[t] = (n < NOUT) ? 1.0f : 0.0f;
    }

    v8f acc[NT] = {};
    const float* xr = x   + (size_t)arow * FEAT;
    const float* ar = agg + (size_t)arow * FEAT;

#pragma unroll 4
    for (int k = 0; k < FEAT; k += 4) {
        v2f ax, aa;
        ax.x = xr[k + kk];
        ax.y = xr[k + kk + 1];
        aa.x = ar[k + kk] * scale;
        aa.y = ar[k + kk + 1] * scale;
        const float* wsr0 = Ws + (size_t)(k + kk)     * NOUT;
        const float* wsr1 = Ws + (size_t)(k + kk + 1) * NOUT;
        const float* wnr0 = Wn + (size_t)(k + kk)     * NOUT;
        const float* wnr1 = Wn + (size_t)(k + kk + 1) * NOUT;
#pragma unroll
        for (int t = 0; t < NT; ++t) {
            v2f bs, bn;
            bs.x = wsr0[ncol[t]] * nmsk[t];
            bs.y = wsr1[ncol[t]] * nmsk[t];
            bn.x = wnr0[ncol[t]] * nmsk[t];
            bn.y = wnr1[ncol[t]] * nmsk[t];
            acc[t] = __builtin_amdgcn_wmma_f32_16x16x4_f32(
                false, ax, false, bs, (short)0, acc[t], false, false);
            acc[t] = __builtin_amdgcn_wmma_f32_16x16x4_f32(
                false, aa, false, bn, (short)0, acc[t], false, false);
        }
    }

    // Epilogue: bias (+ReLU), store per C/D layout.
    const int ncl   = lane & 15;
    const int mhalf = (lane >> 4) * 8;
#pragma unroll
    for (int t = 0; t < NT; ++t) {
        int col = t * 16 + ncl;
        if (col < NOUT) {
            float bv = bias[col];
#pragma unroll
            for (int r = 0; r < 8; ++r) {
                float v = acc[t][r] + bv;
                if (RELU) v = fmaxf(v, 0.0f);
                out[(size_t)(m0 + mhalf + r) * NOUT + col] = v;
            }
        }
    }
}

// ---------------------------------------------------------------- launcher
extern "C" void kernel_launch(void* const* d_in, const int* in_sizes, int n_in,
                              void* d_out, int out_size, void* d_ws, size_t ws_size,
                              hipStream_t stream) {
    const float* features = (const float*)d_in[0];
    const int*   src      = (const int*)  d_in[1];
    const int*   dst      = (const int*)  d_in[2];
    const float* Ws1      = (const float*)d_in[3];
    const float* Wn1      = (const float*)d_in[4];
    const float* b1       = (const float*)d_in[5];
    const float* Ws2      = (const float*)d_in[6];
    const float* Wn2      = (const float*)d_in[7];
    const float* b2       = (const float*)d_in[8];
    const float* Ws3      = (const float*)d_in[9];
    const float* Wn3      = (const float*)d_in[10];
    const float* b3       = (const float*)d_in[11];
    float*       out      = (float*)d_out;

    float* ws      = (float*)d_ws;
    float* inv_deg = ws;                               // N_NODES floats
    float* agg     = inv_deg + N_NODES;                // N_NODES*128
    float* h1      = agg + (size_t)N_NODES * FEAT;     // N_NODES*128
    float* h2      = h1  + (size_t)N_NODES * FEAT;     // N_NODES*128

    const int aggN       = N_NODES * FEAT;             // 12.8M
    const int fillBlocks = (aggN + 255) / 256;
    const int scatBlocks = (N_EDGES * 32) / 256;       // 200000
    const int gemmBlocks = N_NODES / 16;               // 6250

    // degree -> inv_deg
    fill_zero_kernel<<<(N_NODES + 255) / 256, 256, 0, stream>>>(inv_deg, N_NODES);
    degree_kernel<<<(N_EDGES + 255) / 256, 256, 0, stream>>>(dst, inv_deg);
    invdeg_kernel<<<(N_NODES + 255) / 256, 256, 0, stream>>>(inv_deg);

    // layer 1
    fill_zero_kernel<<<fillBlocks, 256, 0, stream>>>(agg, aggN);
    scatter_add_kernel<<<scatBlocks, 256, 0, stream>>>(features, src, dst, agg);
    sage_gemm_kernel<8, true, 128><<<gemmBlocks, 32, 0, stream>>>(
        features, agg, inv_deg, Ws1, Wn1, b1, h1);

    // layer 2
    fill_zero_kernel<<<fillBlocks, 256, 0, stream>>>(agg, aggN);
    scatter_add_kernel<<<scatBlocks, 256, 0, stream>>>(h1, src, dst, agg);
    sage_gemm_kernel<8, true, 128><<<gemmBlocks, 32, 0, stream>>>(
        h1, agg, inv_deg, Ws2, Wn2, b2, h2);

    // layer 3 (40 classes, masked tail tile, no ReLU)
    fill_zero_kernel<<<fillBlocks, 256, 0, stream>>>(agg, aggN);
    scatter_add_kernel<<<scatBlocks, 256, 0, stream>>>(h2, src, dst, agg);
    sage_gemm_kernel<3, false, 40><<<gemmBlocks, 32, 0, stream>>>(
        h2, agg, inv_deg, Ws3, Wn3, b3, out);
}